// Voxelization_63093069578687
// MI455X (gfx1250) — compile-verified
//
#include <hip/hip_runtime.h>
#include <math.h>

typedef __attribute__((ext_vector_type(2))) float v2f;
typedef __attribute__((ext_vector_type(8))) float v8f;

#define GRID_G   64
#define G3       262144      // 64^3 = 2^18
#define BATCH    2
#define MROWS    524288      // BATCH * G3
#define NTILES   32768       // MROWS / 16
#define EPS_NORM 1e-6f
#define EPS_BN   1e-5f
#define W2_STRIDE 136        // 128 + 8: half-wave bank sets disjoint (2*136 % 64 == 16)

static __device__ __forceinline__ v8f wmma_f32_k4(v2f a, v2f b, v8f c) {
  // D = A(16x4,f32) * B(4x16,f32) + C(16x16,f32)
  return __builtin_amdgcn_wmma_f32_16x16x4_f32(
      /*neg_a=*/false, a, /*neg_b=*/false, b,
      /*c_mod=*/(short)0, c, /*reuse_a=*/false, /*reuse_b=*/false);
}

// Stage W2 (64x128 f32, row-major) into LDS with padded row stride, using the
// CDNA5 async global->LDS DMA path (ASYNCcnt). 128 threads * 16 iters * 16B
// covers the matrix exactly; 16B chunks never cross a padded LDS row.
static __device__ __forceinline__ void stage_w2_async(const float* __restrict__ W2,
                                                      unsigned ldsbase, int tid) {
#pragma unroll
  for (int it = 0; it < 16; ++it) {
    int chunk = it * 128 + tid;          // 0..2047
    int krow  = chunk >> 5;              // 0..63
    int cj    = (chunk & 31) << 2;       // 0,4,...,124
    const float* g = W2 + krow * 128 + cj;
    unsigned loff = ldsbase + (unsigned)(((krow * W2_STRIDE + cj) << 2));
    asm volatile("global_load_async_to_lds_b128 %0, %1, off"
                 :: "v"(loff), "v"(g)
                 : "memory");
  }
  asm volatile("s_wait_asynccnt 0x0" ::: "memory");
}

// ---- float atomic min/max via int punning (correct for mixed signs) ----
static __device__ __forceinline__ void atomicMinF(float* addr, float v) {
  if (v >= 0.0f) atomicMin((int*)addr, __float_as_int(v));
  else           atomicMax((unsigned int*)addr, __float_as_uint(v));
}
static __device__ __forceinline__ void atomicMaxF(float* addr, float v) {
  if (v >= 0.0f) atomicMax((int*)addr, __float_as_int(v));
  else           atomicMin((unsigned int*)addr, __float_as_uint(v));
}

// ---------------- K0: init workspace ----------------
__global__ void k_init(float* sums, float* minmax, float* stats, int nsums) {
  int i = blockIdx.x * blockDim.x + threadIdx.x;
  if (i < nsums) sums[i] = 0.0f;
  if (i < 384)   stats[i] = 0.0f;             // stats1(128) + stats2(256), contiguous
  if (i < 6)     minmax[i] = 3.402823e38f;    // mins
  else if (i < 12) minmax[i] = -3.402823e38f; // maxs
}

// ---------------- K1: per-batch per-axis min/max ----------------
__global__ void k_minmax(const float* __restrict__ pc, float* minmax, int N) {
  int b = blockIdx.y;
  int n = blockIdx.x * blockDim.x + threadIdx.x;
  bool act = n < N;
  __shared__ float red[256];
  for (int a = 0; a < 3; ++a) {
    float v = act ? pc[((size_t)b * 3 + a) * N + n] : 0.0f;
    red[threadIdx.x] = act ? v : 3.402823e38f;
    __syncthreads();
    for (int s = 128; s > 0; s >>= 1) {
      if (threadIdx.x < s) red[threadIdx.x] = fminf(red[threadIdx.x], red[threadIdx.x + s]);
      __syncthreads();
    }
    if (threadIdx.x == 0) atomicMinF(&minmax[b * 3 + a], red[0]);
    __syncthreads();
    red[threadIdx.x] = act ? v : -3.402823e38f;
    __syncthreads();
    for (int s = 128; s > 0; s >>= 1) {
      if (threadIdx.x < s) red[threadIdx.x] = fmaxf(red[threadIdx.x], red[threadIdx.x + s]);
      __syncthreads();
    }
    if (threadIdx.x == 0) atomicMaxF(&minmax[6 + b * 3 + a], red[0]);
    __syncthreads();
  }
}

// ---------------- K2: scatter points into voxels ----------------
__global__ void k_scatter(const float* __restrict__ pc, const float* __restrict__ minmax,
                          float* sums, int N) {
  int b = blockIdx.y;
  int n = blockIdx.x * blockDim.x + threadIdx.x;
  if (n >= N) return;
  float x = pc[((size_t)b * 3 + 0) * N + n];
  float y = pc[((size_t)b * 3 + 1) * N + n];
  float z = pc[((size_t)b * 3 + 2) * N + n];
  float mnx = minmax[b * 3 + 0], mny = minmax[b * 3 + 1], mnz = minmax[b * 3 + 2];
  float mxx = minmax[6 + b * 3 + 0], mxy = minmax[6 + b * 3 + 1], mxz = minmax[6 + b * 3 + 2];
  int ix = (int)(((x - mnx) / (mxx - mnx + EPS_NORM)) * (GRID_G - 1));
  int iy = (int)(((y - mny) / (mxy - mny + EPS_NORM)) * (GRID_G - 1));
  int iz = (int)(((z - mnz) / (mxz - mnz + EPS_NORM)) * (GRID_G - 1));
  int id = b * G3 + ix * (GRID_G * GRID_G) + iy * GRID_G + iz;
  atomicAdd(&sums[(size_t)id * 4 + 0], x);
  atomicAdd(&sums[(size_t)id * 4 + 1], y);
  atomicAdd(&sums[(size_t)id * 4 + 2], z);
  atomicAdd(&sums[(size_t)id * 4 + 3], 1.0f);
}

// ---------------- K3: sums -> (mean xyz, density) in place ----------------
__global__ void k_finalize(float* sums, float invN) {
  int i = blockIdx.x * blockDim.x + threadIdx.x;
  if (i >= MROWS) return;
  float cnt = sums[(size_t)i * 4 + 3];
  float inv = (cnt > 0.0f) ? (1.0f / fmaxf(cnt, 1.0f)) : 0.0f;
  sums[(size_t)i * 4 + 0] *= inv;
  sums[(size_t)i * 4 + 1] *= inv;
  sums[(size_t)i * 4 + 2] *= inv;
  sums[(size_t)i * 4 + 3] = cnt * invN;
}

// ---------------- K4: layer-1 pre-activation stats (WMMA f32 K=4) ----------------
__global__ __launch_bounds__(128) void k_pass1(const float* __restrict__ feat,
                                               const float* __restrict__ W1,
                                               const float* __restrict__ b1,
                                               float* stats1) {
  __shared__ float s1[64], ss1[64];
  int tid = threadIdx.x;
  if (tid < 64) { s1[tid] = 0.0f; ss1[tid] = 0.0f; }
  __syncthreads();
  int wave = tid >> 5, lane = tid & 31;
  int lrow = lane & 15;
  int k0 = (lane < 16) ? 0 : 2;
  int tile = blockIdx.x * 4 + wave;
  int row = tile * 16 + lrow;
  v2f a; a.x = feat[(size_t)row * 4 + k0]; a.y = feat[(size_t)row * 4 + k0 + 1];
  for (int c = 0; c < 4; ++c) {
    int col = c * 16 + lrow;
    v2f bm; bm.x = W1[k0 * 64 + col]; bm.y = W1[(k0 + 1) * 64 + col];
    v8f acc = {0.f,0.f,0.f,0.f,0.f,0.f,0.f,0.f};
    acc = wmma_f32_k4(a, bm, acc);
    float bias = b1[col];
    float s = 0.0f, ss = 0.0f;
#pragma unroll
    for (int v = 0; v < 8; ++v) { float y = acc[v] + bias; s += y; ss += y * y; }
    atomicAdd(&s1[col], s);
    atomicAdd(&ss1[col], ss);
  }
  __syncthreads();
  if (tid < 64) { atomicAdd(&stats1[tid], s1[tid]); atomicAdd(&stats1[64 + tid], ss1[tid]); }
}

// ---------------- K5/K7: BN coefficients: h = a*y + c ----------------
__global__ void k_bncoef(const float* __restrict__ stats, const float* __restrict__ gamma,
                         const float* __restrict__ beta, float* ac, int nfeat, float invM) {
  int j = blockIdx.x * blockDim.x + threadIdx.x;
  if (j >= nfeat) return;
  float mu  = stats[j] * invM;
  float var = stats[nfeat + j] * invM - mu * mu;
  float a   = gamma[j] * rsqrtf(var + EPS_BN);
  ac[j]         = a;
  ac[nfeat + j] = beta[j] - mu * a;
}

// ---------------- K6: layer-2 pre-activation stats ----------------
__global__ __launch_bounds__(128) void k_pass2(const float* __restrict__ feat,
                                               const float* __restrict__ W1,
                                               const float* __restrict__ b1,
                                               const float* __restrict__ ac1,
                                               const float* __restrict__ W2,
                                               const float* __restrict__ b2,
                                               float* stats2) {
  __shared__ __align__(16) float w2s[64 * W2_STRIDE];  // staged W2 (padded rows)
  __shared__ float hbuf[4][16][65];                    // per-wave h tile
  __shared__ float s2[128], ss2[128];
  int tid = threadIdx.x;
  if (tid < 128) { s2[tid] = 0.0f; ss2[tid] = 0.0f; }
  stage_w2_async(W2, (unsigned)(uintptr_t)&w2s[0], tid);
  __syncthreads();
  int wave = tid >> 5, lane = tid & 31;
  int lrow = lane & 15;
  int k0 = (lane < 16) ? 0 : 2;
  int mhi = (lane >= 16) ? 8 : 0;
  int tile = blockIdx.x * 4 + wave;
  int row = tile * 16 + lrow;
  v2f a; a.x = feat[(size_t)row * 4 + k0]; a.y = feat[(size_t)row * 4 + k0 + 1];
  // layer 1: y1 -> BN -> ReLU -> LDS
  for (int c = 0; c < 4; ++c) {
    int col = c * 16 + lrow;
    v2f bm; bm.x = W1[k0 * 64 + col]; bm.y = W1[(k0 + 1) * 64 + col];
    v8f acc = {0.f,0.f,0.f,0.f,0.f,0.f,0.f,0.f};
    acc = wmma_f32_k4(a, bm, acc);
    float bias = b1[col], aa = ac1[col], cc0 = ac1[64 + col];
#pragma unroll
    for (int v = 0; v < 8; ++v) {
      float y = acc[v] + bias;
      hbuf[wave][v + mhi][col] = fmaxf(aa * y + cc0, 0.0f);
    }
  }
  __syncthreads();
  // layer 2: y2 = h(16x64) @ W2(64x128), 16 chained K=4 WMMAs per 16-col tile
  for (int cc = 0; cc < 8; ++cc) {
    int col = cc * 16 + lrow;
    v8f acc = {0.f,0.f,0.f,0.f,0.f,0.f,0.f,0.f};
#pragma unroll
    for (int kk = 0; kk < 16; ++kk) {
      int k = 4 * kk + k0;
      v2f av; av.x = hbuf[wave][lrow][k];       av.y = hbuf[wave][lrow][k + 1];
      v2f bv; bv.x = w2s[k * W2_STRIDE + col];  bv.y = w2s[(k + 1) * W2_STRIDE + col];
      acc = wmma_f32_k4(av, bv, acc);
    }
    float bias = b2[col];
    float s = 0.0f, ss = 0.0f;
#pragma unroll
    for (int v = 0; v < 8; ++v) { float y = acc[v] + bias; s += y; ss += y * y; }
    atomicAdd(&s2[col], s);
    atomicAdd(&ss2[col], ss);
  }
  __syncthreads();
  if (tid < 128) { atomicAdd(&stats2[tid], s2[tid]); atomicAdd(&stats2[128 + tid], ss2[tid]); }
}

// ---------------- K8: full forward + transposed store ----------------
__global__ __launch_bounds__(128) void k_pass3(const float* __restrict__ feat,
                                               const float* __restrict__ W1,
                                               const float* __restrict__ b1,
                                               const float* __restrict__ ac1,
                                               const float* __restrict__ W2,
                                               const float* __restrict__ b2,
                                               const float* __restrict__ ac2,
                                               float* __restrict__ out) {
  __shared__ __align__(16) float w2s[64 * W2_STRIDE];
  __shared__ float hbuf[4][16][65];
  int tid = threadIdx.x;
  stage_w2_async(W2, (unsigned)(uintptr_t)&w2s[0], tid);
  __syncthreads();
  int wave = tid >> 5, lane = tid & 31;
  int lrow = lane & 15;
  int k0 = (lane < 16) ? 0 : 2;
  int mhi = (lane >= 16) ? 8 : 0;
  int tile = blockIdx.x * 4 + wave;
  int row = tile * 16 + lrow;
  v2f a; a.x = feat[(size_t)row * 4 + k0]; a.y = feat[(size_t)row * 4 + k0 + 1];
  for (int c = 0; c < 4; ++c) {
    int col = c * 16 + lrow;
    v2f bm; bm.x = W1[k0 * 64 + col]; bm.y = W1[(k0 + 1) * 64 + col];
    v8f acc = {0.f,0.f,0.f,0.f,0.f,0.f,0.f,0.f};
    acc = wmma_f32_k4(a, bm, acc);
    float bias = b1[col], aa = ac1[col], cc0 = ac1[64 + col];
#pragma unroll
    for (int v = 0; v < 8; ++v) {
      float y = acc[v] + bias;
      hbuf[wave][v + mhi][col] = fmaxf(aa * y + cc0, 0.0f);
    }
  }
  __syncthreads();
  for (int cc = 0; cc < 8; ++cc) {
    int col = cc * 16 + lrow;
    v8f acc = {0.f,0.f,0.f,0.f,0.f,0.f,0.f,0.f};
#pragma unroll
    for (int kk = 0; kk < 16; ++kk) {
      int k = 4 * kk + k0;
      v2f av; av.x = hbuf[wave][lrow][k];       av.y = hbuf[wave][lrow][k + 1];
      v2f bv; bv.x = w2s[k * W2_STRIDE + col];  bv.y = w2s[(k + 1) * W2_STRIDE + col];
      acc = wmma_f32_k4(av, bv, acc);
    }
    float bias = b2[col], a2 = ac2[col], c2 = ac2[128 + col];
#pragma unroll
    for (int v = 0; v < 8; ++v) {
      int r = tile * 16 + v + mhi;
      int bidx = r >> 18;            // G3 = 2^18
      int vox  = r & (G3 - 1);
      float y = acc[v] + bias;
      float o = fmaxf(a2 * y + c2, 0.0f);
      out[((size_t)bidx * 128 + col) * (size_t)G3 + vox] = o;
    }
  }
}

extern "C" void kernel_launch(void* const* d_in, const int* in_sizes, int n_in,
                              void* d_out, int out_size, void* d_ws, size_t ws_size,
                              hipStream_t stream) {
  const float* pc  = (const float*)d_in[0];
  const float* W1  = (const float*)d_in[1];
  const float* b1  = (const float*)d_in[2];
  const float* g1  = (const float*)d_in[3];
  const float* be1 = (const float*)d_in[4];
  const float* W2  = (const float*)d_in[5];
  const float* b2  = (const float*)d_in[6];
  const float* g2  = (const float*)d_in[7];
  const float* be2 = (const float*)d_in[8];
  float* out = (float*)d_out;
  float* ws  = (float*)d_ws;

  int N = in_sizes[0] / (BATCH * 3);

  // workspace layout (floats)
  float* sums   = ws;                 // MROWS*4 = 2,097,152
  float* minmax = ws + 2097152;       // 12
  float* stats1 = ws + 2097164;       // 128 (sum|sumsq)  -- contiguous with stats2
  float* stats2 = ws + 2097292;       // 256
  float* ac1    = ws + 2097548;       // 128 (a|c)
  float* ac2    = ws + 2097676;       // 256

  int nsums = MROWS * 4;
  k_init<<<(nsums + 255) / 256, 256, 0, stream>>>(sums, minmax, stats1, nsums);

  dim3 gmm((N + 255) / 256, BATCH);
  k_minmax<<<gmm, 256, 0, stream>>>(pc, minmax, N);
  k_scatter<<<gmm, 256, 0, stream>>>(pc, minmax, sums, N);
  k_finalize<<<(MROWS + 255) / 256, 256, 0, stream>>>(sums, 1.0f / (float)N);

  const float invM = 1.0f / (float)MROWS;
  k_pass1<<<NTILES / 4, 128, 0, stream>>>(sums, W1, b1, stats1);
  k_bncoef<<<1, 64, 0, stream>>>(stats1, g1, be1, ac1, 64, invM);
  k_pass2<<<NTILES / 4, 128, 0, stream>>>(sums, W1, b1, ac1, W2, b2, stats2);
  k_bncoef<<<1, 128, 0, stream>>>(stats2, g2, be2, ac2, 128, invM);
  k_pass3<<<NTILES / 4, 128, 0, stream>>>(sums, W1, b1, ac1, W2, b2, ac2, out);
}